// Attention_84756884619813
// MI455X (gfx1250) — compile-verified
//
#include <hip/hip_runtime.h>
#include <cstdint>

typedef __attribute__((ext_vector_type(2))) float v2f;
typedef __attribute__((ext_vector_type(8))) float v8f;

__device__ __forceinline__ v8f wmma4(v2f a, v2f b, v8f c) {
  // V_WMMA_F32_16X16X4_F32: D = A(16x4) * B(4x16) + C(16x16)
  return __builtin_amdgcn_wmma_f32_16x16x4_f32(
      /*neg_a=*/false, a, /*neg_b=*/false, b,
      /*c_mod=*/(short)0, c, /*reuse_a=*/false, /*reuse_b=*/false);
}

__device__ __forceinline__ int dmin(int a, int b) { return a < b ? a : b; }

// --- CDNA5 async memory->LDS (ASYNCcnt-tracked), inline asm (portable) ------
__device__ __forceinline__ unsigned lds_off32(const void* p) {
  // AS3 generic address: low 32 bits are the wave-relative LDS byte offset
  return (unsigned)(unsigned long long)p;
}
__device__ __forceinline__ void async_ld_b128(unsigned lds_addr, const float* g) {
  asm volatile("global_load_async_to_lds_b128 %0, %1, off"
               :: "v"(lds_addr), "v"((unsigned long long)g) : "memory");
}
#define WAIT_ASYNC(imm) asm volatile("s_wait_asynccnt " imm ::: "memory")

// Problem constants
constexpr int BB    = 16;    // batch
constexpr int NP    = 576;
constexpr int CC    = 1024;  // channels
constexpr int HH    = 16;    // heads
constexpr int HD    = 64;    // head dim
constexpr int TT    = 578;   // tokens incl. memory token
constexpr int NQ    = 577;   // query rows (= keys without memory token)
constexpr int RET   = 410;   // RETAIN
constexpr int LDM   = 592;   // padded leading dim for M / S tiles (37*16)
constexpr float SCALE = 0.125f;                  // hd^-0.5
constexpr float MFAC  = 0.125f / (16.0f * 0.5f); // scale/(H*tau)
constexpr float EPSV  = 1e-6f;

// ---------------------------------------------------------------------------
// K1: qkv = tokens(B*578 x 1024) @ Wqkv(1024 x 3072)
// tile 64x128, 8 waves (2x4), each wave 32x32 via 4 wmma accumulators.
// Tiles staged via global_load_async_to_lds_b128 with double buffering.
// ---------------------------------------------------------------------------
__global__ __launch_bounds__(256)
void qkv_gemm(const float* __restrict__ x, const float* __restrict__ mem,
              const float* __restrict__ W, float* __restrict__ qkv)
{
  const int tid  = threadIdx.x;
  const int lane = tid & 31, wave = tid >> 5;
  const int g = lane >> 4, lr = lane & 15;
  const int m0 = blockIdx.y * 64;    // gridDim.y = 145
  const int n0 = blockIdx.x * 128;   // gridDim.x = 24
  const int m_off = (wave & 1) * 32;
  const int n_off = (wave >> 1) * 32;
  __shared__ float As[2][64][16];
  __shared__ float Bs[2][16][132];

  v8f acc00 = {}, acc01 = {}, acc10 = {}, acc11 = {};
  const int a_row = tid >> 2, a_col = (tid & 3) * 4;
  const int b_kk  = tid >> 4, b_nc  = (tid & 15) * 8;

  const int M = BB * TT;
  // Clamp A row: C row r depends only on A row r, and OOB rows are never
  // stored in the epilogue, so duplicated rows are harmless.
  const int gr = dmin(m0 + a_row, M - 1);
  const int bbv = gr / TT, t = gr % TT;
  const float* arow = (t < NQ) ? (x + ((size_t)bbv * NQ + t) * CC)
                               : (mem + (size_t)bbv * CC);
  const float* brow = W + (size_t)b_kk * 3072 + n0 + b_nc;

  auto stage = [&](int c, int p) {
    const int k0 = c * 16;
    async_ld_b128(lds_off32(&As[p][a_row][a_col]), arow + k0 + a_col);
    async_ld_b128(lds_off32(&Bs[p][b_kk][b_nc]),     brow + (size_t)k0 * 3072);
    async_ld_b128(lds_off32(&Bs[p][b_kk][b_nc + 4]), brow + (size_t)k0 * 3072 + 4);
  };

  stage(0, 0);
  for (int c = 0; c < 64; c++) {
    const int cur = c & 1;
    if (c + 1 < 64) { stage(c + 1, cur ^ 1); WAIT_ASYNC("0x3"); }
    else            { WAIT_ASYNC("0x0"); }
    __syncthreads();
#pragma unroll
    for (int kk = 0; kk < 16; kk += 4) {
      const int ka = kk + 2 * g;
      v2f a0, a1, b0, b1;
      a0.x = As[cur][m_off + lr][ka];        a0.y = As[cur][m_off + lr][ka + 1];
      a1.x = As[cur][m_off + 16 + lr][ka];   a1.y = As[cur][m_off + 16 + lr][ka + 1];
      b0.x = Bs[cur][ka][n_off + lr];        b0.y = Bs[cur][ka + 1][n_off + lr];
      b1.x = Bs[cur][ka][n_off + 16 + lr];   b1.y = Bs[cur][ka + 1][n_off + 16 + lr];
      acc00 = wmma4(a0, b0, acc00);
      acc01 = wmma4(a0, b1, acc01);
      acc10 = wmma4(a1, b0, acc10);
      acc11 = wmma4(a1, b1, acc11);
    }
    __syncthreads();
  }
#pragma unroll
  for (int ms = 0; ms < 2; ms++)
#pragma unroll
    for (int ns = 0; ns < 2; ns++) {
      v8f acc = ms == 0 ? (ns == 0 ? acc00 : acc01) : (ns == 0 ? acc10 : acc11);
#pragma unroll
      for (int r = 0; r < 8; r++) {
        const int row = m0 + m_off + ms * 16 + r + 8 * g;
        const int col = n0 + n_off + ns * 16 + lr;
        if (row < M) qkv[(size_t)row * 3072 + col] = acc[r];
      }
    }
}

// ---------------------------------------------------------------------------
// K2: M[b] = (scale/(H*tau)) * Q[b](577x1024) @ K[b]^T(1024x577)
// B-tile loaded transposed (per-element scatter), with next-chunk prefetch.
// ---------------------------------------------------------------------------
__global__ __launch_bounds__(256)
void score_gemm(const float* __restrict__ qkv, float* __restrict__ accM)
{
  const int tid  = threadIdx.x;
  const int lane = tid & 31, wave = tid >> 5;
  const int g = lane >> 4, lr = lane & 15;
  const int b  = blockIdx.z;
  const int m0 = blockIdx.y * 64;    // gridDim.y = 10
  const int n0 = blockIdx.x * 128;   // gridDim.x = 5
  const int m_off = (wave & 1) * 32, n_off = (wave >> 1) * 32;
  __shared__ float As[64][16];
  __shared__ float Bs[16][132];
  v8f acc00 = {}, acc01 = {}, acc10 = {}, acc11 = {};
  const int a_row = tid >> 2, a_col = (tid & 3) * 4;
  const int b_j = tid & 127, b_k0 = (tid >> 7) * 8;
  const float* base = qkv + (size_t)b * TT * 3072;
  const bool a_ok = (m0 + a_row) < NQ;
  const bool b_ok = (n0 + b_j) < NQ;
  const float* ap = base + (size_t)(m0 + a_row) * 3072 + a_col;
  const float* kp = base + (size_t)(n0 + b_j) * 3072 + 1024 + b_k0;

  for (int k0 = 0; k0 < CC; k0 += 16) {
    float4 av = make_float4(0.f, 0.f, 0.f, 0.f);
    if (a_ok) av = *(const float4*)(ap + k0);
    *(float4*)&As[a_row][a_col] = av;
    float bv[8];
    if (b_ok) {
      float4 t0 = *(const float4*)(kp + k0), t1 = *(const float4*)(kp + k0 + 4);
      bv[0] = t0.x; bv[1] = t0.y; bv[2] = t0.z; bv[3] = t0.w;
      bv[4] = t1.x; bv[5] = t1.y; bv[6] = t1.z; bv[7] = t1.w;
    } else {
#pragma unroll
      for (int i = 0; i < 8; i++) bv[i] = 0.f;
    }
#pragma unroll
    for (int i = 0; i < 8; i++) Bs[b_k0 + i][b_j] = bv[i];
    if (k0 + 16 < CC) {                    // -> global_prefetch_b8
      if (a_ok) __builtin_prefetch(ap + k0 + 16, 0, 3);
      if (b_ok) __builtin_prefetch(kp + k0 + 16, 0, 3);
    }
    __syncthreads();
#pragma unroll
    for (int kk = 0; kk < 16; kk += 4) {
      const int ka = kk + 2 * g;
      v2f a0, a1, b0, b1;
      a0.x = As[m_off + lr][ka];        a0.y = As[m_off + lr][ka + 1];
      a1.x = As[m_off + 16 + lr][ka];   a1.y = As[m_off + 16 + lr][ka + 1];
      b0.x = Bs[ka][n_off + lr];        b0.y = Bs[ka + 1][n_off + lr];
      b1.x = Bs[ka][n_off + 16 + lr];   b1.y = Bs[ka + 1][n_off + 16 + lr];
      acc00 = wmma4(a0, b0, acc00);
      acc01 = wmma4(a0, b1, acc01);
      acc10 = wmma4(a1, b0, acc10);
      acc11 = wmma4(a1, b1, acc11);
    }
    __syncthreads();
  }
#pragma unroll
  for (int ms = 0; ms < 2; ms++)
#pragma unroll
    for (int ns = 0; ns < 2; ns++) {
      v8f acc = ms == 0 ? (ns == 0 ? acc00 : acc01) : (ns == 0 ? acc10 : acc11);
#pragma unroll
      for (int r = 0; r < 8; r++) {
        const int row = m0 + m_off + ms * 16 + r + 8 * g;
        const int col = n0 + n_off + ns * 16 + lr;
        if (row < NQ && col < NQ)
          accM[((size_t)b * NQ + row) * LDM + col] = acc[r] * MFAC;
      }
    }
}

// ---------------------------------------------------------------------------
// K3: attention per (16 q-rows, head, batch); one wave per block.
// Phase1: S = scale * q kT (WMMA over hd=64), V rows prefetched;
// Phase2: masked softmax variant; Phase3: out = P @ V (WMMA over 578 keys).
// ---------------------------------------------------------------------------
__global__ __launch_bounds__(32)
void attn_kernel(const float* __restrict__ qkv, const float* __restrict__ token_mask,
                 float* __restrict__ out_heads)
{
  const int lane = threadIdx.x, g = lane >> 4, lr = lane & 15;
  const int qt = blockIdx.x, h = blockIdx.y, b = blockIdx.z;
  const int q0 = qt * 16;
  __shared__ float S[16][LDM];
  const float* base = qkv + (size_t)b * TT * 3072;
  const int ch = h * HD;

  // Phase 1: scores
  const int trow = dmin(q0 + lr, NQ - 1);
  const float* qrow = base + (size_t)trow * 3072 + ch;
  for (int nt = 0; nt < 37; nt++) {
    const int n0 = nt * 16;
    const int j = n0 + lr;
    const bool jv = j < TT;
    const float* krow = base + (size_t)dmin(j, TT - 1) * 3072 + 1024 + ch;
    if (jv) __builtin_prefetch(base + (size_t)j * 3072 + 2048 + ch, 0, 3); // V row
    v8f acc = {};
#pragma unroll
    for (int k = 0; k < HD; k += 4) {
      const int ka = k + 2 * g;
      v2f a, bv;
      a.x = qrow[ka]; a.y = qrow[ka + 1];
      bv.x = jv ? krow[ka] : 0.f;
      bv.y = jv ? krow[ka + 1] : 0.f;
      acc = wmma4(a, bv, acc);
    }
#pragma unroll
    for (int r = 0; r < 8; r++) S[r + 8 * g][n0 + lr] = acc[r] * SCALE;
  }
  __syncthreads();

  // Phase 2: e = exp(S)*key_mask; P = (e + eps/T)/(sum(e) + eps)
  if (lane < 16) {
    const int row = lane;
    float sum = 0.f;
    for (int j = 0; j < TT; j++) {
      const float km = (j == 0 || j == TT - 1) ? 1.f
                       : token_mask[(size_t)b * NP + (j - 1)];
      const float e = __expf(S[row][j]) * km;
      S[row][j] = e;
      sum += e;
    }
    const float inv = 1.f / (sum + EPSV);
    const float add = EPSV / (float)TT;
    for (int j = 0; j < TT; j++) S[row][j] = (S[row][j] + add) * inv;
    // cols 578..591 remain exactly 0 from phase 1 (zero B fragments)
  }
  __syncthreads();

  // Phase 3: out = P(16x578) @ V(578x64)
  v8f o0 = {}, o1 = {}, o2 = {}, o3 = {};
  for (int j0 = 0; j0 < 580; j0 += 4) {
    const int ka = j0 + 2 * g;
    v2f a;
    a.x = S[lr][ka]; a.y = S[lr][ka + 1];
    const bool v0 = ka < TT, v1 = ka + 1 < TT;
    const float* vr0 = base + (size_t)dmin(ka, TT - 1) * 3072 + 2048 + ch;
    const float* vr1 = base + (size_t)dmin(ka + 1, TT - 1) * 3072 + 2048 + ch;
#pragma unroll
    for (int ns = 0; ns < 4; ns++) {
      const int col = ns * 16 + lr;
      v2f bv;
      bv.x = v0 ? vr0[col] : 0.f;
      bv.y = v1 ? vr1[col] : 0.f;
      v8f& o = ns == 0 ? o0 : (ns == 1 ? o1 : (ns == 2 ? o2 : o3));
      o = wmma4(a, bv, o);
    }
  }
#pragma unroll
  for (int ns = 0; ns < 4; ns++) {
    v8f o = ns == 0 ? o0 : (ns == 1 ? o1 : (ns == 2 ? o2 : o3));
#pragma unroll
    for (int r = 0; r < 8; r++) {
      const int t = q0 + r + 8 * g;
      const int col = ns * 16 + lr;
      if (t < NQ)
        out_heads[((size_t)b * NQ + t) * CC + ch + col] = o[r];
    }
  }
}

// ---------------------------------------------------------------------------
// K4: Mrow normalization in place: M[i][j] <- exp(M)*mask_all[j] / rowsum
// ---------------------------------------------------------------------------
__global__ __launch_bounds__(256)
void mrow_kernel(float* __restrict__ accM, const float* __restrict__ token_mask)
{
  const int i = blockIdx.x, b = blockIdx.y, tid = threadIdx.x;
  __shared__ float red[256];
  float* row = accM + ((size_t)b * NQ + i) * LDM;
  float part = 0.f;
  for (int j = tid; j < NQ; j += 256) {
    const float m = (j == 0) ? 1.f : token_mask[(size_t)b * NP + j - 1];
    part += __expf(row[j]) * m;
  }
  red[tid] = part;
  __syncthreads();
  for (int s = 128; s > 0; s >>= 1) {
    if (tid < s) red[tid] += red[tid + s];
    __syncthreads();
  }
  const float inv = 1.f / red[0];
  for (int j = tid; j < NQ; j += 256) {
    const float m = (j == 0) ? 1.f : token_mask[(size_t)b * NP + j - 1];
    row[j] = __expf(row[j]) * m * inv;
  }
}

// ---------------------------------------------------------------------------
// K5: one power-iteration step: dout[j] = sum_i din[i] * Mrow[i][j]
// ---------------------------------------------------------------------------
__global__ __launch_bounds__(256)
void iter_kernel(const float* __restrict__ Mrow, const float* __restrict__ din,
                 float* __restrict__ dout, int uniform)
{
  const int b = blockIdx.y;
  const int j = blockIdx.x * 256 + threadIdx.x;
  __shared__ float d[NQ];
  for (int i = threadIdx.x; i < NQ; i += 256)
    d[i] = uniform ? (1.f / (float)NQ) : din[(size_t)b * LDM + i];
  __syncthreads();
  if (j < NQ) {
    const float* col = Mrow + (size_t)b * NQ * LDM + j;
    float s = 0.f;
    for (int i = 0; i < NQ; i++) s += d[i] * col[(size_t)i * LDM];
    dout[(size_t)b * LDM + j] = s;
  }
}

// ---------------------------------------------------------------------------
// K6: out = out_heads(9232x1024) @ Wproj(1024x1024) + bproj
// Async double-buffered LDS staging like K1.
// ---------------------------------------------------------------------------
__global__ __launch_bounds__(256)
void proj_gemm(const float* __restrict__ A, const float* __restrict__ W,
               const float* __restrict__ bias, float* __restrict__ out)
{
  const int tid  = threadIdx.x;
  const int lane = tid & 31, wave = tid >> 5;
  const int g = lane >> 4, lr = lane & 15;
  const int m0 = blockIdx.y * 64;    // gridDim.y = 145
  const int n0 = blockIdx.x * 128;   // gridDim.x = 8
  const int m_off = (wave & 1) * 32, n_off = (wave >> 1) * 32;
  __shared__ float As[2][64][16];
  __shared__ float Bs[2][16][132];
  v8f acc00 = {}, acc01 = {}, acc10 = {}, acc11 = {};
  const int a_row = tid >> 2, a_col = (tid & 3) * 4;
  const int b_kk  = tid >> 4, b_nc  = (tid & 15) * 8;
  const int M = BB * NQ;

  const int gr = dmin(m0 + a_row, M - 1);   // clamp (see K1)
  const float* arow = A + (size_t)gr * CC;
  const float* brow = W + (size_t)b_kk * CC + n0 + b_nc;

  auto stage = [&](int c, int p) {
    const int k0 = c * 16;
    async_ld_b128(lds_off32(&As[p][a_row][a_col]), arow + k0 + a_col);
    async_ld_b128(lds_off32(&Bs[p][b_kk][b_nc]),     brow + (size_t)k0 * CC);
    async_ld_b128(lds_off32(&Bs[p][b_kk][b_nc + 4]), brow + (size_t)k0 * CC + 4);
  };

  stage(0, 0);
  for (int c = 0; c < 64; c++) {
    const int cur = c & 1;
    if (c + 1 < 64) { stage(c + 1, cur ^ 1); WAIT_ASYNC("0x3"); }
    else            { WAIT_ASYNC("0x0"); }
    __syncthreads();
#pragma unroll
    for (int kk = 0; kk < 16; kk += 4) {
      const int ka = kk + 2 * g;
      v2f a0, a1, b0, b1;
      a0.x = As[cur][m_off + lr][ka];        a0.y = As[cur][m_off + lr][ka + 1];
      a1.x = As[cur][m_off + 16 + lr][ka];   a1.y = As[cur][m_off + 16 + lr][ka + 1];
      b0.x = Bs[cur][ka][n_off + lr];        b0.y = Bs[cur][ka + 1][n_off + lr];
      b1.x = Bs[cur][ka][n_off + 16 + lr];   b1.y = Bs[cur][ka + 1][n_off + 16 + lr];
      acc00 = wmma4(a0, b0, acc00);
      acc01 = wmma4(a0, b1, acc01);
      acc10 = wmma4(a1, b0, acc10);
      acc11 = wmma4(a1, b1, acc11);
    }
    __syncthreads();
  }
#pragma unroll
  for (int ms = 0; ms < 2; ms++)
#pragma unroll
    for (int ns = 0; ns < 2; ns++) {
      v8f acc = ms == 0 ? (ns == 0 ? acc00 : acc01) : (ns == 0 ? acc10 : acc11);
#pragma unroll
      for (int r = 0; r < 8; r++) {
        const int row = m0 + m_off + ms * 16 + r + 8 * g;
        const int col = n0 + n_off + ns * 16 + lr;
        if (row < M) out[(size_t)row * CC + col] = acc[r] + bias[col];
      }
    }
}

// ---------------------------------------------------------------------------
// K7: top-RETAIN selection among kept tokens (rank with stable tie-break),
// emit next_mask and importance.
// ---------------------------------------------------------------------------
__global__ __launch_bounds__(576)
void select_kernel(const float* __restrict__ token_mask, const float* __restrict__ dist,
                   float* __restrict__ next_mask, float* __restrict__ imp_out)
{
  const int b = blockIdx.x, p = threadIdx.x;
  __shared__ float msk[NP];
  __shared__ float imp[NP];
  msk[p] = token_mask[(size_t)b * NP + p];
  imp[p] = dist[(size_t)b * LDM + 1 + p];  // importance = dist[0, 1:]
  __syncthreads();
  const bool kept = msk[p] > 0.5f;
  int cnt = 0;
  if (kept) {
    const float ip = imp[p];
    for (int q = 0; q < NP; q++) {
      if (msk[q] > 0.5f) {
        const float iq = imp[q];
        if (iq > ip || (iq == ip && q < p)) cnt++;
      }
    }
  }
  next_mask[(size_t)b * NP + p] = (kept && cnt < RET) ? 1.f : 0.f;
  imp_out[(size_t)b * NP + p] = imp[p];
}

// ---------------------------------------------------------------------------
extern "C" void kernel_launch(void* const* d_in, const int* in_sizes, int n_in,
                              void* d_out, int out_size, void* d_ws, size_t ws_size,
                              hipStream_t stream)
{
  (void)in_sizes; (void)n_in; (void)out_size; (void)ws_size;
  const float* x          = (const float*)d_in[0];
  const float* mem        = (const float*)d_in[1];
  const float* token_mask = (const float*)d_in[2];
  const float* Wqkv       = (const float*)d_in[3];
  const float* Wproj      = (const float*)d_in[4];
  const float* bproj      = (const float*)d_in[5];

  char* ws = (char*)d_ws;
  size_t off = 0;
  float* qkv  = (float*)(ws + off); off += (size_t)BB * TT * 3072 * 4;   // 113.6 MB
  float* accM = (float*)(ws + off); off += (size_t)BB * NQ * LDM * 4;    //  21.9 MB
  float* outh = (float*)(ws + off); off += (size_t)BB * NQ * CC * 4;     //  37.8 MB
  float* distA = (float*)(ws + off); off += (size_t)BB * LDM * 4;
  float* distB = (float*)(ws + off);

  float* out       = (float*)d_out;
  float* next_mask = out + (size_t)BB * NQ * CC;
  float* imp_out   = next_mask + (size_t)BB * NP;

  // 1) QKV projection (M=9248, N=3072, K=1024), async LDS staging
  qkv_gemm<<<dim3(24, 145), 256, 0, stream>>>(x, mem, Wqkv, qkv);
  // 2) Importance base matrix: full-channel Q*K^T (per batch)
  score_gemm<<<dim3(5, 10, BB), 256, 0, stream>>>(qkv, accM);
  // 3) Attention per (q-tile, head, batch)
  attn_kernel<<<dim3(37, HH, BB), 32, 0, stream>>>(qkv, token_mask, outh);
  // 4) Row-normalize to Markov matrix
  mrow_kernel<<<dim3(NQ, BB), 256, 0, stream>>>(accM, token_mask);
  // 5) Five power-iteration steps (ping-pong); final result lands in distA
  iter_kernel<<<dim3(3, BB), 256, 0, stream>>>(accM, distB, distA, 1);
  iter_kernel<<<dim3(3, BB), 256, 0, stream>>>(accM, distA, distB, 0);
  iter_kernel<<<dim3(3, BB), 256, 0, stream>>>(accM, distB, distA, 0);
  iter_kernel<<<dim3(3, BB), 256, 0, stream>>>(accM, distA, distB, 0);
  iter_kernel<<<dim3(3, BB), 256, 0, stream>>>(accM, distB, distA, 0);
  // 6) Output projection + bias (M=9232, N=1024, K=1024), async LDS staging
  proj_gemm<<<dim3(8, 145), 256, 0, stream>>>(outh, Wproj, bproj, out);
  // 7) Selection / next_mask / importance outputs
  select_kernel<<<BB, NP, 0, stream>>>(token_mask, distA, next_mask, imp_out);
}